// MyModel_61933428414413
// MI455X (gfx1250) — compile-verified
//
#include <hip/hip_runtime.h>
#include <hip/hip_bf16.h>
#include <stdint.h>

// Latency-bound sequential chain: v <- x .* v, num_iter times, on 4 elements.
// - 4 independent scalar chains live in lanes 0..3 of a single wave32; each
//   loop step is one dependent v_mul_f32 per chain (the latency floor given
//   the reference's strict sequential fp semantics — WMMA per link would be
//   strictly slower: XDL latency + RAW-hazard slots vs 1 dependent VALU mul).
// - The pass-through copy of x -> out[0:4] uses the CDNA5 async LDS DMA path
//   (ASYNCcnt-tracked), issued before the chain so it overlaps the compute.
__global__ __launch_bounds__(32) void chain_mul_kernel(
    const float* __restrict__ x, const float* __restrict__ v,
    const int* __restrict__ niter_p, float* __restrict__ out) {
  __shared__ __align__(16) float xtile[4];
  const int t = (int)threadIdx.x;

  // --- Kick off async DMA: x (16 bytes) -> LDS. One active lane, B128. ---
  if (t == 0) {
    unsigned lds_addr = (unsigned)(size_t)&xtile[0];  // low 32 bits = LDS offset
    asm volatile("global_load_async_to_lds_b128 %0, %1, off"
                 :: "v"(lds_addr), "v"(x)
                 : "memory");
  }

  // --- Latency-bound dependent multiply chain (lanes 0..3 carry real data) ---
  const int n = *niter_p;
  float xv = 1.0f, acc = 1.0f;
  if (t < 4) {
    xv  = x[t];        // x[i][j], row-major, t = i*2 + j
    acc = v[t >> 1];   // broadcast v[i] across j
  }
  // Force the operand loads to resolve here so no s_wait lands in the loop.
  asm volatile("" : "+v"(xv), "+v"(acc));
#pragma unroll 32
  for (int k = 0; k < n; ++k) {
    acc *= xv;         // strict sequential chain; not reassociable
  }

  // --- Drain async load, then async store LDS -> out[0:4] (copy of x) ---
  asm volatile("s_wait_asynccnt 0x0" ::: "memory");
  if (t == 0) {
    unsigned lds_addr = (unsigned)(size_t)&xtile[0];
    asm volatile("global_store_async_from_lds_b128 %0, %1, off"
                 :: "v"(out), "v"(lds_addr)
                 : "memory");
  }

  // --- v_out part of the tuple ---
  if (n == 0) {
    if (t < 2) out[4 + t] = v[t];     // num_iter==0: v stays [2,1]
  } else {
    if (t < 4) out[4 + t] = acc;      // v_out is [2,2]
  }

  // Ensure the async store has left before wave end (S_ENDPGM also waits-idle).
  asm volatile("s_wait_asynccnt 0x0" ::: "memory");
}

extern "C" void kernel_launch(void* const* d_in, const int* in_sizes, int n_in,
                              void* d_out, int out_size, void* d_ws, size_t ws_size,
                              hipStream_t stream) {
  const float* x = (const float*)d_in[0];   // 4 floats
  const float* v = (const float*)d_in[1];   // 2 floats
  const int*   n = (const int*)d_in[2];     // scalar num_iter
  float* out = (float*)d_out;               // 8 floats: [x(4), v_out(4)]
  (void)in_sizes; (void)n_in; (void)out_size; (void)d_ws; (void)ws_size;

  chain_mul_kernel<<<1, 32, 0, stream>>>(x, v, n, out);
}